// HierarchicalSoftmax_40785009443163
// MI455X (gfx1250) — compile-verified
//
#include <hip/hip_runtime.h>
#include <hip/hip_bf16.h>
#include <math.h>

// ---------------- problem constants (from reference) ----------------
#define NHID     512
#define NCLASSES 316          // ceil(100000 / 317)
#define TPC      317
#define BATCHSZ  2048
#define NC_TILES 20           // ceil(316/16)

// output layout (flat float concat in tuple return order)
#define OFF_LOSS    ((size_t)0)
#define OFF_TARGET  ((size_t)1)
#define OFF_TOP     ((size_t)2049)                       // 1 + 2048
#define OFF_BOT     ((size_t)(2049 + 2048*316))          // 649217
#define OFF_TIDX    ((size_t)(649217 + 2048*317))        // 1298433
#define OFF_BIDX    ((size_t)(1298433 + 2048))           // 1300481
#define OFF_RIDX    ((size_t)(1300481 + 2048))           // 1302529
#define OFF_PREDS   ((size_t)(1302529 + 2048))           // 1304577

typedef __attribute__((ext_vector_type(2))) float v2f;
typedef __attribute__((ext_vector_type(8))) float v8f;

// ---------------------------------------------------------------
// Top GEMM: logits[2048,316] = X[2048,512] @ W[512,316] + b
// One wave (32 threads) per 16x16 C tile, V_WMMA_F32_16X16X4_F32,
// K-loop of 128 WMMA ops. fp32 end to end (matches reference).
// A frag (16x4 f32): lanes 0-15 hold K=koff..koff+1, lanes 16-31 K=koff+2..3.
// B frag (4x16 f32): lane n holds B[k][n] in v0, B[k+1][n] in v1 (halves split K).
// C/D (16x16 f32): vgpr r = row r (lanes0-15) / row r+8 (lanes16-31), col = lane%16.
// ---------------------------------------------------------------
__global__ __launch_bounds__(32) void top_gemm_wmma(
    const float* __restrict__ X, const float* __restrict__ W,
    const float* __restrict__ bias, float* __restrict__ out)
{
    const int mt   = blockIdx.x;        // 0..127
    const int nt   = blockIdx.y;        // 0..19
    const int lane = threadIdx.x;       // 0..31
    const int half = lane >> 4;         // 0 / 1
    const int l16  = lane & 15;

    const int m    = mt * 16 + l16;
    const int n    = nt * 16 + l16;
    const int nn   = (n < NCLASSES) ? n : (NCLASSES - 1);  // clamp B col loads
    const int koff = half * 2;

    const float* xrow = X + (size_t)m * NHID;

    v8f c = {};
    for (int k = 0; k < NHID; k += 4) {
        v2f a = *(const v2f*)(xrow + k + koff);            // contiguous 64-bit load
        v2f b;
        b.x = W[(size_t)(k + koff)     * NCLASSES + nn];
        b.y = W[(size_t)(k + koff + 1) * NCLASSES + nn];
        c = __builtin_amdgcn_wmma_f32_16x16x4_f32(
                /*neg_a=*/false, a, /*neg_b=*/false, b,
                /*c_mod=*/(short)0, c, /*reuse_a=*/false, /*reuse_b=*/false);
    }

    if (n < NCLASSES) {
        const float bv = bias[n];
        #pragma unroll
        for (int r = 0; r < 8; ++r) {
            const int row = mt * 16 + r + half * 8;
            out[(size_t)row * NCLASSES + n] = c[r] + bv;
        }
    }
}

// ---------------------------------------------------------------
// Bottom logits: per sample b, class c = labels[b]/317:
//   logit[b,t] = sum_d X[b,d] * Wb[c,d,t] + bb[c,t]
// One block per sample; X row staged in LDS; threads stride t so the
// Wb walk is coalesced at every d step. L2 reuse across same-class rows.
// ---------------------------------------------------------------
__global__ __launch_bounds__(256) void bottom_logits_kernel(
    const float* __restrict__ X, const int* __restrict__ labels,
    const float* __restrict__ Wb, const float* __restrict__ bb,
    float* __restrict__ out)
{
    __shared__ float sx[NHID];
    const int b   = blockIdx.x;
    const int tid = threadIdx.x;

    for (int d = tid; d < NHID; d += 256) sx[d] = X[(size_t)b * NHID + d];
    __syncthreads();

    const int c = labels[b] / TPC;
    const float* W = Wb + (size_t)c * NHID * TPC;

    for (int t = tid; t < TPC; t += 256) {
        float acc = bb[(size_t)c * TPC + t];
        const float* wp = W + t;
        #pragma unroll 8
        for (int d = 0; d < NHID; ++d)
            acc = fmaf(sx[d], wp[(size_t)d * TPC], acc);
        out[(size_t)b * TPC + t] = acc;
    }
}

// ---------------------------------------------------------------
// In-place row softmax + argmax (argmax of logits == argmax of probs).
// One block (128 threads) per row.
// ---------------------------------------------------------------
__global__ __launch_bounds__(128) void softmax_argmax_kernel(
    float* __restrict__ data, int N, float* __restrict__ idx_out)
{
    __shared__ float sval[128];
    __shared__ int   sidx[128];
    __shared__ float ssum[128];

    const int b   = blockIdx.x;
    const int tid = threadIdx.x;
    float* row = data + (size_t)b * N;

    float mv = -3.4e38f; int mi = 0;
    for (int i = tid; i < N; i += 128) {
        float v = row[i];
        if (v > mv) { mv = v; mi = i; }
    }
    sval[tid] = mv; sidx[tid] = mi;
    __syncthreads();
    for (int s = 64; s > 0; s >>= 1) {
        if (tid < s) {
            float ov = sval[tid + s]; int oi = sidx[tid + s];
            if (ov > sval[tid] || (ov == sval[tid] && oi < sidx[tid])) {
                sval[tid] = ov; sidx[tid] = oi;
            }
        }
        __syncthreads();
    }
    const float rmax = sval[0];
    const int   amax = sidx[0];

    float lsum = 0.f;
    for (int i = tid; i < N; i += 128) {
        float e = __expf(row[i] - rmax);
        row[i] = e;
        lsum += e;
    }
    ssum[tid] = lsum;
    __syncthreads();
    for (int s = 64; s > 0; s >>= 1) {
        if (tid < s) ssum[tid] += ssum[tid + s];
        __syncthreads();
    }
    const float inv = 1.0f / ssum[0];
    for (int i = tid; i < N; i += 128) row[i] *= inv;

    if (tid == 0) idx_out[b] = (float)amax;
}

// ---------------------------------------------------------------
// Per-sample target prob + real index.
// ---------------------------------------------------------------
__global__ __launch_bounds__(256) void finalize_kernel(
    const int* __restrict__ labels,
    const float* __restrict__ top, const float* __restrict__ bot,
    const float* __restrict__ tidx, const float* __restrict__ bidx,
    float* __restrict__ target_out, float* __restrict__ ridx_out)
{
    const int b = blockIdx.x * blockDim.x + threadIdx.x;
    if (b >= BATCHSZ) return;
    const int lab = labels[b];
    const int pt = lab / TPC;
    const int pb = lab - pt * TPC;
    target_out[b] = top[(size_t)b * NCLASSES + pt] * bot[(size_t)b * TPC + pb];
    ridx_out[b]   = tidx[b] * (float)TPC + bidx[b];
}

// ---------------------------------------------------------------
// loss = -mean(log(target_probs)); single-block reduction.
// ---------------------------------------------------------------
__global__ __launch_bounds__(256) void loss_kernel(
    const float* __restrict__ target, float* __restrict__ loss_out)
{
    __shared__ float ssum[256];
    const int tid = threadIdx.x;
    float s = 0.f;
    for (int i = tid; i < BATCHSZ; i += 256) s += __logf(target[i]);
    ssum[tid] = s;
    __syncthreads();
    for (int k = 128; k > 0; k >>= 1) {
        if (tid < k) ssum[tid] += ssum[tid + k];
        __syncthreads();
    }
    if (tid == 0) loss_out[0] = -ssum[0] / (float)BATCHSZ;
}

// ---------------------------------------------------------------
// preds[b, c*317+t] = top[b,c] * bot[b,t]  (820 MB of writes: the
// dominant cost). Rows staged in LDS; b32 stores fully coalesced.
// ---------------------------------------------------------------
__global__ __launch_bounds__(256) void preds_kernel(
    const float* __restrict__ top, const float* __restrict__ bot,
    float* __restrict__ preds)
{
    __shared__ float st[NCLASSES];
    __shared__ float sb[TPC];
    const int b   = blockIdx.x;
    const int tid = threadIdx.x;

    for (int i = tid; i < NCLASSES; i += 256) st[i] = top[(size_t)b * NCLASSES + i];
    for (int i = tid; i < TPC;      i += 256) sb[i] = bot[(size_t)b * TPC + i];
    __syncthreads();

    const float v0 = sb[tid];
    const bool  has2 = (tid + 256) < TPC;
    const float v1 = has2 ? sb[tid + 256] : 0.f;

    float* row = preds + (size_t)b * NCLASSES * TPC;
    for (int c = 0; c < NCLASSES; ++c) {
        const float tp = st[c];
        float* rp = row + (size_t)c * TPC;
        rp[tid] = tp * v0;
        if (has2) rp[tid + 256] = tp * v1;
    }
}

// ---------------------------------------------------------------
extern "C" void kernel_launch(void* const* d_in, const int* in_sizes, int n_in,
                              void* d_out, int out_size, void* d_ws, size_t ws_size,
                              hipStream_t stream) {
    (void)in_sizes; (void)n_in; (void)d_ws; (void)ws_size; (void)out_size;

    const float* X    = (const float*)d_in[0];   // [2048,512]
    const int*   lab  = (const int*)  d_in[1];   // [2048]
    // d_in[2] = ntokens_per_class scalar (compile-time constant TPC)
    const float* topW = (const float*)d_in[3];   // [512,316]
    const float* topB = (const float*)d_in[4];   // [316]
    const float* botW = (const float*)d_in[5];   // [316,512,317]
    const float* botB = (const float*)d_in[6];   // [316,317]

    float* out      = (float*)d_out;
    float* loss     = out + OFF_LOSS;
    float* target   = out + OFF_TARGET;
    float* top_p    = out + OFF_TOP;     // logits -> probs in place
    float* bot_p    = out + OFF_BOT;     // logits -> probs in place
    float* tidx     = out + OFF_TIDX;
    float* bidx     = out + OFF_BIDX;
    float* ridx     = out + OFF_RIDX;
    float* preds    = out + OFF_PREDS;

    // 1) top logits via fp32 WMMA
    top_gemm_wmma<<<dim3(BATCHSZ / 16, NC_TILES), 32, 0, stream>>>(X, topW, topB, top_p);

    // 2) bottom logits (gathered per-sample weight matrices)
    bottom_logits_kernel<<<BATCHSZ, 256, 0, stream>>>(X, lab, botW, botB, bot_p);

    // 3/4) softmax + argmax (in place)
    softmax_argmax_kernel<<<BATCHSZ, 128, 0, stream>>>(top_p, NCLASSES, tidx);
    softmax_argmax_kernel<<<BATCHSZ, 128, 0, stream>>>(bot_p, TPC, bidx);

    // 5) target probs + real index
    finalize_kernel<<<(BATCHSZ + 255) / 256, 256, 0, stream>>>(lab, top_p, bot_p,
                                                               tidx, bidx, target, ridx);

    // 6) loss
    loss_kernel<<<1, 256, 0, stream>>>(target, loss);

    // 7) preds outer product (dominant: ~820 MB of coalesced stores)
    preds_kernel<<<BATCHSZ, 256, 0, stream>>>(top_p, bot_p, preds);
}